// IdDeformConv2d_56865366999363
// MI455X (gfx1250) — compile-verified
//
#include <hip/hip_runtime.h>
#include <math.h>

// Problem constants (fixed by the reference setup)
#define NPTS 8000
#define AUXN 4000
#define CCH  256
#define OCH  256
#define KKN  9
#define PN   2
#define RH   64
#define RW   64
#define KC   (KKN * CCH)      // 2304
#define PAD_IDX (NPTS + AUXN) // 12000

typedef __attribute__((ext_vector_type(2))) float v2f;
typedef __attribute__((ext_vector_type(8))) float v8f;

// ---------------------------------------------------------------------------
// Kernel 1: per-point prep. One wave32 per point.
//   - 63 dot products (off 36, pw 18, mw 9) against the point's 256 features
//   - lanes 0..8 then build the 72 (gid, weight) pairs for their k
// ---------------------------------------------------------------------------
__global__ __launch_bounds__(256)
void prep_kernel(const float* __restrict__ in_core,
                 const float* __restrict__ off_w, const float* __restrict__ off_b,
                 const float* __restrict__ pw_w,  const float* __restrict__ pw_b,
                 const float* __restrict__ mw_w,  const float* __restrict__ mw_b,
                 const int*   __restrict__ id_map,
                 const int*   __restrict__ roi_ids,
                 const int*   __restrict__ pos_ids,
                 int*   __restrict__ g_gid,
                 float* __restrict__ g_wt)
{
    __shared__ float lin_s[8][64];
    const int wave = threadIdx.x >> 5;
    const int lane = threadIdx.x & 31;
    const int n    = blockIdx.x * 8 + wave;   // grid = 1000 blocks -> n < 8000 always

    // per-lane slice of the 256 features (stride-32)
    float f[8];
    const float* fp = in_core + (size_t)n * CCH + lane;
#pragma unroll
    for (int j = 0; j < 8; ++j) f[j] = fp[j * 32];

    // 63 GEMV rows, butterfly wave reduction
    for (int r = 0; r < 63; ++r) {
        const float* wr;
        float bv;
        if (r < 36)      { wr = off_w + r * CCH;        bv = off_b[r];      }
        else if (r < 54) { wr = pw_w + (r - 36) * CCH;  bv = pw_b[r - 36];  }
        else             { wr = mw_w + (r - 54) * CCH;  bv = mw_b[r - 54];  }
        float s = 0.f;
#pragma unroll
        for (int j = 0; j < 8; ++j) s = fmaf(f[j], wr[lane + j * 32], s);
#pragma unroll
        for (int m = 16; m >= 1; m >>= 1) s += __shfl_xor(s, m, 32);
        if (lane == 0) lin_s[wave][r] = s + bv;
    }
    __syncthreads();

    if (lane < KKN) {
        const int k   = lane;
        const int roi = roi_ids[n];
        const float px = (float)pos_ids[n * 2 + 0];
        const float py = (float)pos_ids[n * 2 + 1];

        // softmax over P=2
        const float e0 = lin_s[wave][36 + k * 2 + 0];
        const float e1 = lin_s[wave][36 + k * 2 + 1];
        const float mx = fmaxf(e0, e1);
        const float s0 = __expf(e0 - mx), s1 = __expf(e1 - mx);
        const float inv = 1.f / (s0 + s1);
        const float pwv[2] = { s0 * inv, s1 * inv };
        // sigmoid mask
        const float mwv = 1.f / (1.f + __expf(-lin_s[wave][54 + k]));

        const int* rmap = id_map + roi * (RH * RW);
        const int gx[4] = {0, 1, 0, 1};
        const int gy[4] = {0, 0, 1, 1};
#pragma unroll
        for (int p = 0; p < PN; ++p) {
            const float ox = px + lin_s[wave][k * 4 + p * 2 + 0];
            const float oy = py + lin_s[wave][k * 4 + p * 2 + 1];
            const float fx = floorf(ox), fy = floorf(oy);
            const int bx = (int)fx, by = (int)fy;
            const float dx = ox - fx, dy = oy - fy;
            // reference: w[g] = delta[x_ids[g]] * (1-delta)[y_ids[g]],
            //            x_ids=[1,0,1,0], y_ids=[1,1,0,0]
            const float w4[4] = { dy * (1.f - dy), dx * (1.f - dy),
                                  dy * (1.f - dx), dx * (1.f - dx) };
#pragma unroll
            for (int g = 0; g < 4; ++g) {
                const int cx = bx + gx[g];
                const int cy = by + gy[g];
                int gid;
                if (cx < 0 || cy < 0 || cx >= RW || cy >= RH) gid = PAD_IDX;
                else gid = rmap[cy * RW + cx];
                const int idx = n * 72 + k * 8 + p * 4 + g;
                g_gid[idx] = gid;
                g_wt[idx]  = mwv * pwv[p] * w4[g];
            }
        }
    }
}

// ---------------------------------------------------------------------------
// Kernel 2: transpose weight (256 x 2304) -> wT (2304 x 256) so the GEMM's
// B-fragment loads are lane-coalesced. One-time 2.36 MB.
// ---------------------------------------------------------------------------
__global__ __launch_bounds__(256)
void transpose_kernel(const float* __restrict__ w, float* __restrict__ wT)
{
    const int kc = blockIdx.x;      // 0..2303
    const int o  = threadIdx.x;     // 0..255
    wT[kc * OCH + o] = w[(size_t)o * KC + kc];
}

// ---------------------------------------------------------------------------
// Kernel 3: gather + weighted accumulate -> sampled (8000 x 2304).
// One 64-thread block per point, float4 per thread (global_load_b128 bursts);
// (gid, w) broadcast via LDS so branches stay uniform; all_feats (12.3 MB)
// lives in L2.
// ---------------------------------------------------------------------------
__global__ __launch_bounds__(64)
void gather_kernel(const float* __restrict__ in_core,
                   const float* __restrict__ aux,
                   const int*   __restrict__ g_gid,
                   const float* __restrict__ g_wt,
                   float* __restrict__ sampled)
{
    __shared__ int   sg[72];
    __shared__ float sw[72];
    const int n = blockIdx.x;
    const int t = threadIdx.x;          // 0..63
    if (t < 64)     { sg[t] = g_gid[n * 72 + t];      sw[t] = g_wt[n * 72 + t]; }
    if (t < 8)      { sg[64 + t] = g_gid[n * 72 + 64 + t];
                      sw[64 + t] = g_wt[n * 72 + 64 + t]; }
    __syncthreads();

    const int c4 = t * 4;               // channel base for this thread
#pragma unroll
    for (int k = 0; k < KKN; ++k) {
        float acc0 = 0.f, acc1 = 0.f, acc2 = 0.f, acc3 = 0.f;
#pragma unroll
        for (int j = 0; j < 8; ++j) {
            const int   gid = sg[k * 8 + j];
            const float w   = sw[k * 8 + j];
            float4 v = make_float4(0.f, 0.f, 0.f, 0.f);
            if (gid < NPTS)
                v = *(const float4*)(in_core + (size_t)gid * CCH + c4);
            else if (gid < PAD_IDX)
                v = *(const float4*)(aux + (size_t)(gid - NPTS) * CCH + c4);
            acc0 = fmaf(w, v.x, acc0);
            acc1 = fmaf(w, v.y, acc1);
            acc2 = fmaf(w, v.z, acc2);
            acc3 = fmaf(w, v.w, acc3);
        }
        float4* dst = (float4*)(sampled + (size_t)n * KC + k * CCH + c4);
        *dst = make_float4(acc0, acc1, acc2, acc3);
    }
}

// ---------------------------------------------------------------------------
// Kernel 4: out (8000 x 256) = sampled (8000 x 2304) @ weight.T + bias
// using V_WMMA_F32_16X16X4_F32 (f32 matrix pipe; preserves reference fp32
// numerics over the K=2304 reduction).
//
// One wave per 64x64 output tile: 4 A-fragments x 4 B-fragments -> 16 WMMAs
// per K=4 step, 16 FLOP/byte of fragment traffic. Grid: 125 blocks x 128
// threads; the 4 waves of a block are the 4 column groups of the same 64
// rows, so their A-fragment loads coincide (WGP$/L2 hits).
// All waves fully populated -> EXEC all-ones for every WMMA.
//
// A-fragment (16x4 f32): lanes 0-15 hold M=lane with K=kc..kc+1 in v[0..1];
// lanes 16-31 hold M=lane-16 with K=kc+2..kc+3 (per-lane contiguous float2).
// B-fragment (4x16 f32): lanes 0-15 hold N=lane, K=kc..kc+1; lanes 16-31
// hold K=kc+2..kc+3 (loaded from the transposed weight wT).
// ---------------------------------------------------------------------------
__global__ __launch_bounds__(128)
void gemm_kernel(const float* __restrict__ sampled,
                 const float* __restrict__ wT,
                 const float* __restrict__ bias,
                 float* __restrict__ out)
{
    const int wave   = threadIdx.x >> 5;      // 0..3  -> column group
    const int lane   = threadIdx.x & 31;
    const int m_base = blockIdx.x * 64;       // 125 blocks * 64 rows = 8000
    const int o_base = wave * 64;
    const int hi     = lane >> 4;             // K half select (0: K0/K1, 1: K2/K3)
    const int lo     = lane & 15;             // M (for A) / N (for B)

    const v2f* __restrict__ arow[4];
#pragma unroll
    for (int i = 0; i < 4; ++i)
        arow[i] = (const v2f*)(sampled + (size_t)(m_base + i * 16 + lo) * KC);

    v8f acc[4][4] = {};

    for (int kc = 0; kc < KC; kc += 4) {
        const float* bp = wT + (size_t)(kc + 2 * hi) * OCH + o_base + lo;
        // prefetch next K-chunk of the B stream (global_prefetch_b8)
        __builtin_prefetch(bp + 4 * OCH, 0, 1);

        v2f a[4];
#pragma unroll
        for (int i = 0; i < 4; ++i) a[i] = arow[i][(kc >> 1) + hi];

#pragma unroll
        for (int j = 0; j < 4; ++j) {
            v2f b;
            b[0] = bp[j * 16];          // K = kc + 2*hi
            b[1] = bp[j * 16 + OCH];    // K = kc + 2*hi + 1
#pragma unroll
            for (int i = 0; i < 4; ++i) {
                acc[i][j] = __builtin_amdgcn_wmma_f32_16x16x4_f32(
                    /*neg_a=*/false, a[i], /*neg_b=*/false, b,
                    /*c_mod=*/(short)0, acc[i][j],
                    /*reuse_a=*/false, /*reuse_b=*/false);
            }
        }
    }

    // C/D layout: VGPR r -> M = sub_base + r + 8*hi, N = o_base + j*16 + lo
#pragma unroll
    for (int j = 0; j < 4; ++j) {
        const int o = o_base + j * 16 + lo;
        const float bv = bias[o];
#pragma unroll
        for (int i = 0; i < 4; ++i) {
            const int ms = m_base + i * 16 + 8 * hi;
#pragma unroll
            for (int r = 0; r < 8; ++r) {
                out[(size_t)(ms + r) * OCH + o] = acc[i][j][r] + bv;
            }
        }
    }
}

// ---------------------------------------------------------------------------
// Host-side launch. Workspace layout (bytes, all 256B aligned):
//   [0,        2304000)  g_gid   : int  [8000*72]
//   [2304000,  4608000)  g_wt    : f32  [8000*72]
//   [4608000,  6967296)  wT      : f32  [2304*256]
//   [6967296, 80695296)  sampled : f32  [8000*2304]
// ---------------------------------------------------------------------------
extern "C" void kernel_launch(void* const* d_in, const int* in_sizes, int n_in,
                              void* d_out, int out_size, void* d_ws, size_t ws_size,
                              hipStream_t stream)
{
    const float* in_core = (const float*)d_in[0];
    const float* aux     = (const float*)d_in[1];
    const float* off_w   = (const float*)d_in[2];
    const float* off_b   = (const float*)d_in[3];
    const float* pw_w    = (const float*)d_in[4];
    const float* pw_b    = (const float*)d_in[5];
    const float* mw_w    = (const float*)d_in[6];
    const float* mw_b    = (const float*)d_in[7];
    const float* weight  = (const float*)d_in[8];
    const float* bias    = (const float*)d_in[9];
    const int*   id_map  = (const int*)d_in[10];
    const int*   roi_ids = (const int*)d_in[11];
    const int*   pos_ids = (const int*)d_in[12];
    float* out = (float*)d_out;

    char* ws = (char*)d_ws;
    int*   g_gid   = (int*)(ws);
    float* g_wt    = (float*)(ws + 2304000);
    float* wT      = (float*)(ws + 4608000);
    float* sampled = (float*)(ws + 6967296);

    prep_kernel<<<NPTS / 8, 256, 0, stream>>>(in_core, off_w, off_b, pw_w, pw_b,
                                              mw_w, mw_b, id_map, roi_ids, pos_ids,
                                              g_gid, g_wt);
    transpose_kernel<<<KC, 256, 0, stream>>>(weight, wT);
    gather_kernel<<<NPTS, 64, 0, stream>>>(in_core, aux, g_gid, g_wt, sampled);
    // 125 row-blocks x 4 col-group waves = 500 waves of 64x64 tiles
    gemm_kernel<<<125, 128, 0, stream>>>(sampled, wT, bias, out);
}